// TensorCircuit_44985487458700
// MI455X (gfx1250) — compile-verified
//
#include <hip/hip_runtime.h>

// ---------------- circuit constants (match reference) ----------------
#define V_ 128
#define M_ 256
#define C_ 64
#define B_ 256
#define L_ 4
#define N_ 32768
#define K_ 4
#define A_ 2
#define E_ (N_ * K_)
#define NUM_INPUT (V_ * M_)
#define BASE_ (1 + NUM_INPUT)
#define TOTAL_ROWS (BASE_ + L_ * N_)
#define NEG_INF (-3.4e38f)

typedef __attribute__((ext_vector_type(2))) float v2f;
typedef __attribute__((ext_vector_type(8))) float v8f;

__device__ __forceinline__ float4 ld4(const float* p) { return *(const float4*)p; }
__device__ __forceinline__ void st4(float* p, float4 v) { *(float4*)p = v; }

// ============================================================================
// Kernel 1: input layer.  Block = 4 input nodes; 64 lanes/node, 4 batch/lane.
// node_buf[1+n][b] = logits[n][cat(b)] - logsumexp_c(logits[n][:]); row0 = 0.
// ============================================================================
__global__ __launch_bounds__(256) void input_layer_kernel(
    const int* __restrict__ inputs,           // [B, V]
    const float* __restrict__ input_logits,   // [NUM_INPUT, C]
    float* __restrict__ nodeBuf)              // [TOTAL_ROWS, B]
{
    __shared__ float s_logit[4][C_];
    __shared__ float s_logZ[4];
    const int tid  = threadIdx.x;
    const int sub  = tid >> 6;        // node within block
    const int lane = tid & 63;
    const int n    = blockIdx.x * 4 + sub;

    s_logit[sub][lane] = input_logits[(size_t)n * C_ + lane];
    __syncthreads();

    if (lane == 0) {                  // one thread per node: logsumexp over 64
        float m = NEG_INF;
        for (int c = 0; c < C_; ++c) m = fmaxf(m, s_logit[sub][c]);
        float s = 0.f;
        for (int c = 0; c < C_; ++c) s += __expf(s_logit[sub][c] - m);
        s_logZ[sub] = m + __logf(s);
    }
    __syncthreads();

    const float lz  = s_logZ[sub];
    const int   vid = n >> 8;         // n / M_
    const int   b0  = lane * 4;
    float4 out;
    out.x = s_logit[sub][inputs[(b0 + 0) * V_ + vid]] - lz;
    out.y = s_logit[sub][inputs[(b0 + 1) * V_ + vid]] - lz;
    out.z = s_logit[sub][inputs[(b0 + 2) * V_ + vid]] - lz;
    out.w = s_logit[sub][inputs[(b0 + 3) * V_ + vid]] - lz;
    st4(nodeBuf + (size_t)(1 + n) * B_ + b0, out);

    if (blockIdx.x == 0 && sub == 0) {          // dummy node row 0 = 0.0
        float4 z = {0.f, 0.f, 0.f, 0.f};
        st4(nodeBuf + b0, z);
    }
}

// ============================================================================
// Kernel 2: fused ProdLayer+SumLayer.  Block = 4 sum nodes; 64 lanes/node,
// 4 batch/lane (float4 b128 loads -> 1KB coalesced row reads, L2-resident).
// out[n][b] = logsumexp_k( nb[ch0]+nb[ch1] + sl[n][k] ) - logsumexp_k(sl[n][:])
// ============================================================================
__global__ __launch_bounds__(256) void layer_kernel(
    const int* __restrict__ childIds,    // this layer: [E, A] ints
    const float* __restrict__ sumLogits, // this layer: [N, K]
    float* __restrict__ nodeBuf,
    int dstBase)
{
    __shared__ int   s_ch[4][8];   // 4 nodes * (K*A) child ids
    __shared__ float s_sl[4][4];   // 4 nodes * K edge logits
    const int tid = threadIdx.x;
    if (tid < 32) ((int*)s_ch)[tid]   = childIds[(size_t)blockIdx.x * 32 + tid];
    if (tid < 16) ((float*)s_sl)[tid] = sumLogits[(size_t)blockIdx.x * 16 + tid];
    __syncthreads();

    const int sub  = tid >> 6;
    const int lane = tid & 63;
    const int n    = blockIdx.x * 4 + sub;
    const int b0   = lane * 4;

    float4 t[K_];
#pragma unroll
    for (int k = 0; k < K_; ++k) {
        const int c0 = s_ch[sub][2 * k];
        const int c1 = s_ch[sub][2 * k + 1];
        float4 x0 = ld4(nodeBuf + (size_t)c0 * B_ + b0);
        float4 x1 = ld4(nodeBuf + (size_t)c1 * B_ + b0);
        const float w = s_sl[sub][k];
        t[k].x = x0.x + x1.x + w;
        t[k].y = x0.y + x1.y + w;
        t[k].z = x0.z + x1.z + w;
        t[k].w = x0.w + x1.w + w;
    }
    float4 m = t[0];
#pragma unroll
    for (int k = 1; k < K_; ++k) {
        m.x = fmaxf(m.x, t[k].x); m.y = fmaxf(m.y, t[k].y);
        m.z = fmaxf(m.z, t[k].z); m.w = fmaxf(m.w, t[k].w);
    }
    float4 s = {0.f, 0.f, 0.f, 0.f};
#pragma unroll
    for (int k = 0; k < K_; ++k) {
        s.x += __expf(t[k].x - m.x); s.y += __expf(t[k].y - m.y);
        s.z += __expf(t[k].z - m.z); s.w += __expf(t[k].w - m.w);
    }
    // logsumexp of the 4 edge logits (normalizer of log_softmax)
    const float w0 = s_sl[sub][0], w1 = s_sl[sub][1], w2 = s_sl[sub][2], w3 = s_sl[sub][3];
    const float wm = fmaxf(fmaxf(w0, w1), fmaxf(w2, w3));
    const float zs = wm + __logf(__expf(w0 - wm) + __expf(w1 - wm) +
                                 __expf(w2 - wm) + __expf(w3 - wm));
    float4 o;
    o.x = m.x + __logf(s.x) - zs;
    o.y = m.y + __logf(s.y) - zs;
    o.z = m.z + __logf(s.z) - zs;
    o.w = m.w + __logf(s.w) - zs;
    st4(nodeBuf + (size_t)(dstBase + n) * B_ + b0, o);
}

// ============================================================================
// Root stage 1: partial column max of (x[r][b] + rl[r]) over 256-row chunks.
// ============================================================================
__global__ __launch_bounds__(256) void root_max_kernel(
    const float* __restrict__ lastLayer,     // [N, B]
    const float* __restrict__ root_logits,   // [N]
    float* __restrict__ pmax)                // [128, B]
{
    const int tid = threadIdx.x, i = blockIdx.x;
    float m = NEG_INF;
    for (int j = 0; j < 256; ++j) {
        const int r = i * 256 + j;
        m = fmaxf(m, lastLayer[(size_t)r * B_ + tid] + root_logits[r]);
    }
    pmax[(size_t)i * B_ + tid] = m;
}

// ============================================================================
// Root stage 2: zroot = logsumexp(root_logits); mx[b] = max over chunks.
// ============================================================================
__global__ __launch_bounds__(256) void root_mid_kernel(
    const float* __restrict__ root_logits,
    const float* __restrict__ pmax,
    float* __restrict__ mx,        // [B]
    float* __restrict__ zroot)     // [1]
{
    __shared__ float red[256];
    const int tid = threadIdx.x;
    float lm = NEG_INF;
    for (int j = 0; j < N_ / B_; ++j) lm = fmaxf(lm, root_logits[j * B_ + tid]);
    red[tid] = lm; __syncthreads();
    for (int off = 128; off > 0; off >>= 1) {
        if (tid < off) red[tid] = fmaxf(red[tid], red[tid + off]);
        __syncthreads();
    }
    const float rmax = red[0]; __syncthreads();
    float ls = 0.f;
    for (int j = 0; j < N_ / B_; ++j) ls += __expf(root_logits[j * B_ + tid] - rmax);
    red[tid] = ls; __syncthreads();
    for (int off = 128; off > 0; off >>= 1) {
        if (tid < off) red[tid] += red[tid + off];
        __syncthreads();
    }
    if (tid == 0) zroot[0] = rmax + __logf(red[0]);

    float m = NEG_INF;
    for (int i = 0; i < 128; ++i) m = fmaxf(m, pmax[i * B_ + tid]);
    mx[tid] = m;
}

// ============================================================================
// Root stage 3: WMMA column-sum.  psum[i][b] = sum_{r in chunk i} exp(x+rl-mx).
// Block stages 4 rows x 256 cols of exp-values in LDS; each wave owns two
// 16-col tiles and feeds V_WMMA_F32_16X16X4_F32 with A = ones(16x4), so the
// matrix unit performs exact f32 4-row column sums, accumulated in C/D regs.
// B layout (mirror of 16x4 A): lane L -> col L%16, {K=k0, K=k0+1}, k0=2*(L/16).
// D layout: lane L VGPR0 = D[8*(L/16)][L%16]; all rows equal -> lanes 0..15.
// ============================================================================
#define SP_PAD 16
__global__ __launch_bounds__(256) void root_sum_kernel(
    const float* __restrict__ lastLayer,     // [N, B]
    const float* __restrict__ root_logits,   // [N]
    const float* __restrict__ mx,            // [B]
    float* __restrict__ psum)                // [128, B]
{
    __shared__ float s_p[4][B_ + SP_PAD];
    const int tid    = threadIdx.x;
    const int waveId = tid >> 5;
    const int L      = tid & 31;
    const int i      = blockIdx.x;
    const float myMx = mx[tid];
    const int col = (waveId << 5) + (L & 15);    // first 16-col tile
    const int k0  = (L >> 4) << 1;

    v8f acc0 = {0.f, 0.f, 0.f, 0.f, 0.f, 0.f, 0.f, 0.f};
    v8f acc1 = {0.f, 0.f, 0.f, 0.f, 0.f, 0.f, 0.f, 0.f};
    const v2f ones = {1.0f, 1.0f};

    for (int step = 0; step < 64; ++step) {
        const int r0 = i * 256 + step * 4;
#pragma unroll
        for (int j = 0; j < 4; ++j) {
            const float x = lastLayer[(size_t)(r0 + j) * B_ + tid];
            s_p[j][tid] = __expf(x + root_logits[r0 + j] - myMx);  // <= 1.0, exact in f32
        }
        __syncthreads();
        v2f bf0, bf1;
        bf0[0] = s_p[k0][col];      bf0[1] = s_p[k0 + 1][col];
        bf1[0] = s_p[k0][col + 16]; bf1[1] = s_p[k0 + 1][col + 16];
        acc0 = __builtin_amdgcn_wmma_f32_16x16x4_f32(false, ones, false, bf0,
                                                     (short)0, acc0, false, false);
        acc1 = __builtin_amdgcn_wmma_f32_16x16x4_f32(false, ones, false, bf1,
                                                     (short)0, acc1, false, false);
        __syncthreads();
    }
    if (L < 16) {
        psum[(size_t)i * B_ + col]      = acc0[0];
        psum[(size_t)i * B_ + col + 16] = acc1[0];
    }
}

// ============================================================================
// Root stage 4: lls[b] = mx[b] + log(sum_i psum[i][b]) - zroot.
// ============================================================================
__global__ __launch_bounds__(256) void root_final_kernel(
    const float* __restrict__ psum, const float* __restrict__ mx,
    const float* __restrict__ zroot, float* __restrict__ out)
{
    const int tid = threadIdx.x;
    float s = 0.f;
    for (int i = 0; i < 128; ++i) s += psum[i * B_ + tid];
    out[tid] = mx[tid] + __logf(s) - zroot[0];
}

// ============================================================================
extern "C" void kernel_launch(void* const* d_in, const int* in_sizes, int n_in,
                              void* d_out, int out_size, void* d_ws, size_t ws_size,
                              hipStream_t stream) {
    const int*   inputs       = (const int*)  d_in[0];   // [B, V] int32
    const float* input_logits = (const float*)d_in[1];   // [NUM_INPUT, C]
    const float* sum_logits   = (const float*)d_in[2];   // [L, N, K]
    const float* root_logits  = (const float*)d_in[3];   // [N]
    const int*   prod_child   = (const int*)  d_in[4];   // [L, E, A] int32

    float* ws = (float*)d_ws;
    float* nodeBuf = ws;
    size_t off = (size_t)TOTAL_ROWS * B_;                // ~168 MB, L2-resident
    float* pmax  = ws + off; off += 128 * B_;
    float* psum  = ws + off; off += 128 * B_;
    float* mx    = ws + off; off += B_;
    float* zroot = ws + off; off += 1;

    input_layer_kernel<<<NUM_INPUT / 4, 256, 0, stream>>>(inputs, input_logits, nodeBuf);

    for (int l = 0; l < L_; ++l) {
        layer_kernel<<<N_ / 4, 256, 0, stream>>>(
            prod_child + (size_t)l * E_ * A_,
            sum_logits + (size_t)l * N_ * K_,
            nodeBuf, BASE_ + l * N_);
    }

    const float* lastLayer = nodeBuf + (size_t)(BASE_ + (L_ - 1) * N_) * B_;
    root_max_kernel  <<<128, 256, 0, stream>>>(lastLayer, root_logits, pmax);
    root_mid_kernel  <<<1,   256, 0, stream>>>(root_logits, pmax, mx, zroot);
    root_sum_kernel  <<<128, 256, 0, stream>>>(lastLayer, root_logits, mx, psum);
    root_final_kernel<<<1,   256, 0, stream>>>(psum, mx, zroot, (float*)d_out);
}